// HashSIREN_88029649698982
// MI455X (gfx1250) — compile-verified
//
#include <hip/hip_runtime.h>

// HashGrid + SIREN fused kernel for gfx1250 (MI455X).
// 12-level hash grid encode (96MB table -> L2-resident on 192MB L2) +
// 24->16->16->16->1 SIREN MLP on V_WMMA_F32_16X16X4_F32.
// Each wave32 owns 16 points (M=16). Wave-private LDS tiles + in-order LDS
// means no inter-stage barriers: waves decouple through gather latency.

#define TABLE_SIZE (1u << 20)
#define PRIME_Y 2654435761u

typedef float v2f __attribute__((ext_vector_type(2)));
typedef float v8f __attribute__((ext_vector_type(8)));

__global__ __launch_bounds__(256) void hash_siren_kernel(
    const float* __restrict__ coords,
    const float* __restrict__ table,
    const float* __restrict__ W0, const float* __restrict__ b0,
    const float* __restrict__ W1, const float* __restrict__ b1,
    const float* __restrict__ W2, const float* __restrict__ b2,
    const float* __restrict__ W3, const float* __restrict__ b3,
    float* __restrict__ out, int N)
{
    // Block-shared weight staging (read once per block, reused by 8 waves).
    // W0/b0 pre-scaled by omega0=300 so sin() consumes the WMMA accumulator
    // directly.
    __shared__ float sW0[24 * 16];
    __shared__ float sW1[16 * 16];
    __shared__ float sW2[16 * 16];
    __shared__ float sB0[16], sB1[16], sB2[16], sW3[16];
    __shared__ float sB3;
    // Per-wave staging tiles. sEnc: 16 points x 24 feats, padded stride 25
    // (gcd(25,64)=1 -> conflict-free column reads). sH: 16x16, stride 17.
    __shared__ float sEnc[8][16 * 25];
    __shared__ float sH[8][16 * 17];

    const int tid  = threadIdx.x;
    const int wave = tid >> 5;
    const int lane = tid & 31;
    const int p    = lane & 15;   // point (row M) owned by this lane
    const int half = lane >> 4;   // WMMA A/B fragment half (K pair select)

    for (int i = tid; i < 24 * 16; i += 256) sW0[i] = 300.0f * W0[i];
    for (int i = tid; i < 16 * 16; i += 256) { sW1[i] = W1[i]; sW2[i] = W2[i]; }
    if (tid < 16) {
        sB0[tid] = 300.0f * b0[tid];
        sB1[tid] = b1[tid];
        sB2[tid] = b2[tid];
        sW3[tid] = W3[tid];
    }
    if (tid == 0) sB3 = b3[0];
    __syncthreads();   // only block-wide barrier: weights visible to all waves

    const int base = blockIdx.x * 128 + wave * 16;
    int pg = base + p; if (pg >= N) pg = N - 1;
    const float cx = coords[2 * pg + 0];
    const float cy = coords[2 * pg + 1];

    // ---- multi-resolution hash-grid encoding -------------------------------
    // Lanes 0-15: levels 0..5 (dense indexing, (res+1)^2 <= 2^20).
    // Lanes 16-31: levels 6..11 (spatial hash).
    // Phase 1: issue all 24 independent float2 gathers back-to-back to
    // maximize outstanding LOADcnt (L2 latency hiding), then interpolate.
    {
        const int lvl_base = half * 6;
        float2 t[6][4];
        float  wxa[6], wya[6];

        #pragma unroll
        for (int l = 0; l < 6; ++l) {
            const int   lvl  = lvl_base + l;
            const int   resi = 16 << lvl;
            const float res  = (float)resi;
            const float px = cx * res, py = cy * res;
            const float fx = floorf(px), fy = floorf(py);
            wxa[l] = px - fx;
            wya[l] = py - fy;
            const int ix0 = (int)fx, iy0 = (int)fy;
            const float2* tbl = ((const float2*)table) + (size_t)lvl * TABLE_SIZE;
            #pragma unroll
            for (int c4 = 0; c4 < 4; ++c4) {
                const int dx = c4 & 1, dy = c4 >> 1;
                const int ix = ix0 + dx, iy = iy0 + dy;
                uint32_t idx;
                if (half == 0)
                    idx = (uint32_t)(ix + iy * (resi + 1));
                else
                    idx = (((uint32_t)ix) ^ (((uint32_t)iy) * PRIME_Y)) & (TABLE_SIZE - 1u);
                t[l][c4] = tbl[idx];           // global_load_b64, all independent
            }
        }

        float* encRow = &sEnc[wave][p * 25];
        #pragma unroll
        for (int l = 0; l < 6; ++l) {
            const int lvl = lvl_base + l;
            const float wx = wxa[l], wy = wya[l];
            const float ox = 1.f - wx, oy = 1.f - wy;
            const float w00 = ox * oy, w10 = wx * oy, w01 = ox * wy, w11 = wx * wy;
            float a0, a1;
            a0 = t[l][0].x * w00; a1 = t[l][0].y * w00;
            a0 = fmaf(t[l][1].x, w10, a0); a1 = fmaf(t[l][1].y, w10, a1);
            a0 = fmaf(t[l][2].x, w01, a0); a1 = fmaf(t[l][2].y, w01, a1);
            a0 = fmaf(t[l][3].x, w11, a0); a1 = fmaf(t[l][3].y, w11, a1);
            encRow[2 * lvl + 0] = a0;
            encRow[2 * lvl + 1] = a1;
        }
    }
    // No barrier: sEnc[wave] is wave-private; LDS ops from a wave are in-order.

    // ---- layer 0: [16x24] @ [24x16] + b0, sin(300x). 6 chained f32 WMMAs ---
    v8f acc;
    {
        const float bb = sB0[p];          // bias depends only on N = p
        #pragma unroll
        for (int r = 0; r < 8; ++r) acc[r] = bb;
    }
    {
        const float* enc = sEnc[wave];
        #pragma unroll
        for (int s = 0; s < 6; ++s) {
            const int k0 = 4 * s;
            v2f A, B;
            // A (16x4 f32): vgpr0 = K = k0 + 2*half, vgpr1 = K+1, row M = p
            A.x = enc[p * 25 + k0 + 2 * half + 0];
            A.y = enc[p * 25 + k0 + 2 * half + 1];
            // B (4x16 f32): row K striped across lanes, N = p
            B.x = sW0[(k0 + 2 * half + 0) * 16 + p];
            B.y = sW0[(k0 + 2 * half + 1) * 16 + p];
            acc = __builtin_amdgcn_wmma_f32_16x16x4_f32(
                false, A, false, B, (short)0, acc, false, false);
        }
        // D layout: vgpr r, lane L -> (M = r + 8*half, N = p). Transpose via LDS.
        float* h0 = sH[wave];
        #pragma unroll
        for (int r = 0; r < 8; ++r)
            h0[(r + 8 * half) * 17 + p] = __sinf(acc[r]);   // omega folded into W0/b0
    }

    // ---- layer 1: 16->16, sin. reads sH, writes sEnc (stride 17) -----------
    {
        const float* hin = sH[wave];
        const float bb = sB1[p];
        #pragma unroll
        for (int r = 0; r < 8; ++r) acc[r] = bb;
        #pragma unroll
        for (int s = 0; s < 4; ++s) {
            const int k0 = 4 * s;
            v2f A, B;
            A.x = hin[p * 17 + k0 + 2 * half + 0];
            A.y = hin[p * 17 + k0 + 2 * half + 1];
            B.x = sW1[(k0 + 2 * half + 0) * 16 + p];
            B.y = sW1[(k0 + 2 * half + 1) * 16 + p];
            acc = __builtin_amdgcn_wmma_f32_16x16x4_f32(
                false, A, false, B, (short)0, acc, false, false);
        }
        float* hout = sEnc[wave];
        #pragma unroll
        for (int r = 0; r < 8; ++r)
            hout[(r + 8 * half) * 17 + p] = __sinf(acc[r]);
    }

    // ---- layer 2: 16->16, sin. reads sEnc, writes sH -----------------------
    {
        const float* hin = sEnc[wave];
        const float bb = sB2[p];
        #pragma unroll
        for (int r = 0; r < 8; ++r) acc[r] = bb;
        #pragma unroll
        for (int s = 0; s < 4; ++s) {
            const int k0 = 4 * s;
            v2f A, B;
            A.x = hin[p * 17 + k0 + 2 * half + 0];
            A.y = hin[p * 17 + k0 + 2 * half + 1];
            B.x = sW2[(k0 + 2 * half + 0) * 16 + p];
            B.y = sW2[(k0 + 2 * half + 1) * 16 + p];
            acc = __builtin_amdgcn_wmma_f32_16x16x4_f32(
                false, A, false, B, (short)0, acc, false, false);
        }
        float* hout = sH[wave];
        #pragma unroll
        for (int r = 0; r < 8; ++r)
            hout[(r + 8 * half) * 17 + p] = __sinf(acc[r]);
    }

    // ---- layer 3: 16->1 dot product + b3, one point per lane (lanes 0-15) --
    if (half == 0) {
        const float* hr = &sH[wave][p * 17];
        float o = sB3;
        #pragma unroll
        for (int n = 0; n < 16; ++n)
            o = fmaf(hr[n], sW3[n], o);
        const int pgo = base + p;
        if (pgo < N) out[pgo] = o;
    }
}

extern "C" void kernel_launch(void* const* d_in, const int* in_sizes, int n_in,
                              void* d_out, int out_size, void* d_ws, size_t ws_size,
                              hipStream_t stream) {
    const float* coords = (const float*)d_in[0];
    const float* table  = (const float*)d_in[1];
    const float* W0 = (const float*)d_in[2];
    const float* b0 = (const float*)d_in[3];
    const float* W1 = (const float*)d_in[4];
    const float* b1 = (const float*)d_in[5];
    const float* W2 = (const float*)d_in[6];
    const float* b2 = (const float*)d_in[7];
    const float* W3 = (const float*)d_in[8];
    const float* b3 = (const float*)d_in[9];
    float* out = (float*)d_out;

    const int N = in_sizes[0] / 2;          // coords is [N,2]
    const int blocks = (N + 127) / 128;     // 128 points per 256-thread block
    hash_siren_kernel<<<blocks, 256, 0, stream>>>(
        coords, table, W0, b0, W1, b1, W2, b2, W3, b3, out, N);
}